// GCPLLoss_18554258719481
// MI455X (gfx1250) — compile-verified
//
#include <hip/hip_runtime.h>
#include <cmath>

typedef float v2f __attribute__((ext_vector_type(2)));
typedef float v8f __attribute__((ext_vector_type(8)));

#define EPSV   1e-6f
#define GAMMAV 0.1f
#define TAOV   10.0f
#define BV     1.0f
#define LAMBDAV 0.1f

__device__ __forceinline__ float gfun(float z) {
    // beta = 1: z > 10 ? z : log(1 + exp(z))
    return (z > 10.0f) ? z : log1pf(expf(z));
}

// Kernel 0: q[d] = feature[d] - EPS  (so diff = p - q)
__global__ void gcpl_prep_q(const float* __restrict__ feature,
                            float* __restrict__ q, int D) {
    int i = blockIdx.x * blockDim.x + threadIdx.x;
    if (i < D) q[i] = feature[i] - EPSV;
}

// Kernel 1: per-prototype squared distances via WMMA Gram trick.
// Each wave owns a 16-row tile: rows 0..14 = prototypes, row 15 = q.
// acc = sum_k A_chunk * A_chunk^T  (f32 A and B register layouts coincide,
// so SRC0==SRC1 gives the Gram matrix). Then
//   dist2[i] = G[i,i] - 2*G[i,15] + G[15,15].
__global__ void gcpl_dist2_wmma(const float* __restrict__ protos,
                                const float* __restrict__ q,
                                float* __restrict__ dist2,
                                int CP, int D) {
    __shared__ float lds[8 * 256];                 // 8 waves x (8 VGPR x 32 lanes)
    const int lane = threadIdx.x & 31;
    const int wave = threadIdx.x >> 5;
    const int tile = blockIdx.x * 8 + wave;

    const int r    = lane & 15;                    // matrix row 0..15
    const int half = lane >> 4;                    // K-pair select (+0 / +2)
    const long g   = (long)tile * 15 + r;
    const bool useQ = (r == 15) || (g >= (long)CP);
    const float* rowp = useQ ? q : (protos + g * (long)D);
    rowp += half * 2;

    v8f acc0 = {};
    v8f acc1 = {};
    for (int kb = 0; kb < D; kb += 32) {
        __builtin_prefetch(rowp + kb + 512, 0, 1); // global_prefetch_b8 ahead of stream
        #pragma unroll
        for (int k0 = 0; k0 < 32; k0 += 8) {
            v2f a0 = *(const v2f*)(rowp + kb + k0);     // global_load_b64
            v2f a1 = *(const v2f*)(rowp + kb + k0 + 4); // global_load_b64
            acc0 = __builtin_amdgcn_wmma_f32_16x16x4_f32(
                false, a0, false, a0, (short)0, acc0, false, false);
            acc1 = __builtin_amdgcn_wmma_f32_16x16x4_f32(
                false, a1, false, a1, (short)0, acc1, false, false);
        }
    }
    v8f acc = acc0 + acc1;

    float* w = &lds[wave * 256];
    #pragma unroll
    for (int v = 0; v < 8; ++v) w[v * 32 + lane] = acc[v];
    __syncthreads();

    if (lane < 15) {
        const int i = lane;
        // C/D layout: VGPR v, lanes 0-15 -> M=v ; lanes 16-31 -> M=v+8
        float diag = (i < 8) ? w[i * 32 + i]  : w[(i - 8) * 32 + (i + 16)];
        float c15  = (i < 8) ? w[i * 32 + 15] : w[(i - 8) * 32 + 31];
        float qq   = w[7 * 32 + 31];                  // G[15,15] = ||q||^2
        float d2   = fmaxf(diag - 2.0f * c15 + qq, 0.0f);
        long gg = (long)tile * 15 + i;
        if (gg < (long)CP) dist2[gg] = d2;
    }
}

// Kernel 2: per-class min distance and exp-sum (deterministic tree reduction).
__global__ void gcpl_class_reduce(const float* __restrict__ dist2,
                                  float* __restrict__ dmin,
                                  float* __restrict__ esum, int P) {
    __shared__ float smin[64];
    __shared__ float ssum[64];
    const int c = blockIdx.x;
    const int t = threadIdx.x;
    float d2 = dist2[(long)c * P + t];
    smin[t] = sqrtf(d2);
    ssum[t] = expf(-GAMMAV * d2);
    __syncthreads();
    #pragma unroll
    for (int s = 32; s > 0; s >>= 1) {
        if (t < s) {
            smin[t] = fminf(smin[t], smin[t + s]);
            ssum[t] += ssum[t + s];
        }
        __syncthreads();
    }
    if (t == 0) { dmin[c] = smin[0]; esum[c] = ssum[0]; }
}

// Kernel 3: scalar loss assembly.
__global__ void gcpl_final(const float* __restrict__ dmin,
                           const float* __restrict__ esum,
                           const int* __restrict__ labelp,
                           float* __restrict__ out, int C) {
    __shared__ float s1[1024];
    __shared__ float s2[1024];
    const int t = threadIdx.x;
    const int label = *labelp;
    float e = 0.0f, pw = 0.0f;
    if (t < C) {
        e = esum[t];
        float dm  = dmin[t];
        float dm2 = dm * dm;
        if (t == label) pw = gfun(BV - (TAOV - dm2));   // closest same-class
        else            pw = gfun(BV + (TAOV - dm2));   // other classes
    }
    s1[t] = e; s2[t] = pw;
    __syncthreads();
    #pragma unroll
    for (int s = 512; s > 0; s >>= 1) {
        if (t < s) { s1[t] += s1[t + s]; s2[t] += s2[t + s]; }
        __syncthreads();
    }
    if (t == 0) {
        float one  = s1[0];
        float prob = esum[label] / one;
        float loss = -logf(prob) + LAMBDAV * s2[0];
        out[0] = loss;
        out[1] = dmin[label];
    }
}

extern "C" void kernel_launch(void* const* d_in, const int* in_sizes, int n_in,
                              void* d_out, int out_size, void* d_ws, size_t ws_size,
                              hipStream_t stream) {
    const float* feature = (const float*)d_in[0];   // [1, D]
    const float* protos  = (const float*)d_in[1];   // [C, P, D]
    const int*   label   = (const int*)d_in[2];     // scalar
    float* out = (float*)d_out;                     // {loss, min_distance}

    const int D  = in_sizes[0];                     // 1024
    const int CP = in_sizes[1] / D;                 // 64000
    const int P  = 64;
    const int C  = CP / P;                          // 1000

    float* w     = (float*)d_ws;
    float* q     = w;                               // D floats
    float* dist2 = q + D;                           // CP floats
    float* dmin  = dist2 + CP;                      // C floats
    float* esum  = dmin + C;                        // C floats

    gcpl_prep_q<<<(D + 255) / 256, 256, 0, stream>>>(feature, q, D);

    const int tiles  = (CP + 14) / 15;              // 15 prototype rows per tile
    const int blocks = (tiles + 7) / 8;             // 8 waves (tiles) per block
    gcpl_dist2_wmma<<<blocks, 256, 0, stream>>>(protos, q, dist2, CP, D);

    gcpl_class_reduce<<<C, P, 0, stream>>>(dist2, dmin, esum, P);
    gcpl_final<<<1, 1024, 0, stream>>>(dmin, esum, label, out, C);
}